// DeformableConv_50551765074110
// MI455X (gfx1250) — compile-verified
//
#include <hip/hip_runtime.h>
#include <math.h>

typedef __attribute__((ext_vector_type(2))) float v2f;
typedef __attribute__((ext_vector_type(4))) float v4f;
typedef __attribute__((ext_vector_type(8))) float v8f;
typedef __attribute__((ext_vector_type(4))) int   v4i;

#define GLOBAL_AS __attribute__((address_space(1)))
#define LDS_AS    __attribute__((address_space(3)))

#if __has_builtin(__builtin_amdgcn_global_load_async_to_lds_b32) && \
    __has_builtin(__builtin_amdgcn_global_load_async_to_lds_b128)
#define HAS_ASYNC_LDS 1
#else
#define HAS_ASYNC_LDS 0
#endif

#define B_N   4
#define H_N   128
#define W_N   128
#define C_N   32
#define KT    25          // 5x5 taps
#define G_N   2           // deform groups
#define F_N   64          // output filters
#define OH_N  120
#define OW_N  120
#define OFFC  100         // K*2*G offset channels
#define OCP   112         // OFFC padded to 7 WMMA n-tiles
#define WSTR  34          // LDS stride (floats) for weight panels: conflict-free b64 reads
#define ASTR  36          // LDS stride (floats) for A panels: 16B-aligned, conflict-free

// ---------------------------------------------------------------------------
// Stage 1: dense dilated 5x5 conv producing the 100 offset channels.
// GEMM per wave: M=16 pixels, K=800 (25 taps x 32 ch, 4 at a time), N=112 (7 tiles).
// fp32 WMMA (V_WMMA_F32_16X16X4_F32) keeps full reference precision.
// Weight transpose + A panel staged via GLOBAL_LOAD_ASYNC_TO_LDS (per-lane
// scattered LDS destinations), tracked on ASYNCcnt.
// ---------------------------------------------------------------------------
__global__ __launch_bounds__(256) void offset_conv_wmma(
    const float* __restrict__ vol, const float* __restrict__ w_off,
    const float* __restrict__ b_off, float* __restrict__ offs)
{
    __shared__ float wLds[OCP * WSTR];       // [ocPad][c], transposed per tap
    __shared__ float aLds[8][16 * ASTR];     // per-wave 16x32 A panel

    const int tid  = threadIdx.x;
    const int wave = tid >> 5;
    const int lane = tid & 31;
    const int by   = blockIdx.x;             // b*OH + y  (one output row per block)
    const int b    = by / OH_N;
    const int y    = by - b * OH_N;
    const int x0   = wave * 16;              // wave's 16-pixel M-tile
    const int p    = lane >> 1;              // staging: pixel index (2 lanes/pixel)
    const int h    = lane & 1;               // staging: channel half (16 ch each)
    int xp = x0 + p; if (xp > OW_N - 1) xp = OW_N - 1;   // clamp edge tile (branchless)

    const int col = lane & 15;               // A row m == B col n for this lane
    const int cb  = (lane >> 4) * 2;         // K sub-offset (ISA f32 A layout: K=0/2, K=1/3)

    // Zero the padded oc rows [100,112) once; never overwritten by per-tap staging.
    for (int i = tid; i < (OCP - OFFC) * C_N; i += 256) {
        const int oc = OFFC + (i >> 5), c = i & 31;
        wLds[oc * WSTR + c] = 0.0f;
    }

    v8f acc[7] = {};

    for (int k = 0; k < KT; ++k) {
        __syncthreads();                     // WAR on wLds vs previous iteration
        // Transposing gather: w_off[(k,c),oc] -> wLds[oc][c]
#if HAS_ASYNC_LDS
        for (int i = tid; i < OFFC * C_N; i += 256) {
            const int oc = i >> 5, c = i & 31;
            __builtin_amdgcn_global_load_async_to_lds_b32(
                (GLOBAL_AS int*)(w_off + (k * C_N + c) * OFFC + oc),
                (LDS_AS int*)&wLds[oc * WSTR + c], 0, 0);
        }
#else
        for (int i = tid; i < OFFC * C_N; i += 256) {
            const int oc = i >> 5, c = i & 31;
            wLds[oc * WSTR + c] = w_off[(k * C_N + c) * OFFC + oc];
        }
#endif
        // A panel: direct dilated tap read (no offsets in stage 1) — pure copy
        const int ky = k / 5, kx = k - 5 * ky;
        const float* src = vol +
            (((b * H_N + (y + 2 * ky)) * W_N + (xp + 2 * kx)) * C_N + h * 16);
        v4f* ap = (v4f*)&aLds[wave][p * ASTR + h * 16];
#if HAS_ASYNC_LDS
        __builtin_amdgcn_global_load_async_to_lds_b128(
            (GLOBAL_AS v4i*)src, (LDS_AS v4i*)ap, 0, 0);
        __builtin_amdgcn_global_load_async_to_lds_b128(
            (GLOBAL_AS v4i*)src, (LDS_AS v4i*)ap, 16, 0);
        __builtin_amdgcn_global_load_async_to_lds_b128(
            (GLOBAL_AS v4i*)src, (LDS_AS v4i*)ap, 32, 0);
        __builtin_amdgcn_global_load_async_to_lds_b128(
            (GLOBAL_AS v4i*)src, (LDS_AS v4i*)ap, 48, 0);
#else
        #pragma unroll
        for (int j = 0; j < 4; ++j) ap[j] = ((const v4f*)src)[j];
#endif
        asm volatile("s_wait_asynccnt 0x0" ::: "memory");  // no-op if no async ops
        __syncthreads();                     // wLds + aLds visible to all frag readers

        const float* aRow = &aLds[wave][col * ASTR];
        #pragma unroll
        for (int c0 = 0; c0 < C_N; c0 += 4) {
            v2f a = *(const v2f*)(aRow + c0 + cb);
            #pragma unroll
            for (int t = 0; t < 7; ++t) {
                v2f bb = *(const v2f*)&wLds[(t * 16 + col) * WSTR + c0 + cb];
                acc[t] = __builtin_amdgcn_wmma_f32_16x16x4_f32(
                    false, a, false, bb, (short)0, acc[t], false, false);
            }
        }
    }

    // Epilogue: D layout — VGPR i: lanes0-15 M=i, lanes16-31 M=8+i; N = lane&15
    const int mh = (lane >> 4) * 8;
    for (int t = 0; t < 7; ++t) {
        const int oc = t * 16 + col;
        if (oc >= OFFC) continue;
        const float bias = b_off[oc];
        #pragma unroll
        for (int i = 0; i < 8; ++i) {
            const int xm = x0 + mh + i;
            if (xm < OW_N)
                offs[((b * OH_N + y) * OW_N + xm) * OFFC + oc] = acc[t][i] + bias;
        }
    }
}

// ---------------------------------------------------------------------------
// Stage 2: bilinear gather + grouped einsum via fp32 WMMA.
// Per wave: M=16 pixels; per tap/group build sampled A panel (16x32), then
// 8 K-steps x 2 N-tiles per group. Volume is L2-resident (8.4MB << 192MB L2).
// Weight transpose staged via async-to-LDS; bilinear panel stays on VALU+DS.
// ---------------------------------------------------------------------------
__global__ __launch_bounds__(256) void deform_conv_wmma(
    const float* __restrict__ vol, const float* __restrict__ offs,
    const float* __restrict__ w_dcn, const float* __restrict__ b_dcn,
    float* __restrict__ out)
{
    __shared__ float wLds[G_N * 32 * WSTR];  // [g][f][c], transposed per tap
    __shared__ float aLds[8][16 * ASTR];

    const int tid  = threadIdx.x;
    const int wave = tid >> 5;
    const int lane = tid & 31;
    const int by   = blockIdx.x;
    const int b    = by / OH_N;
    const int y    = by - b * OH_N;
    const int x0   = wave * 16;
    const int p    = lane >> 1;
    const int h    = lane & 1;
    int xp = x0 + p; if (xp > OW_N - 1) xp = OW_N - 1;

    const int col = lane & 15;
    const int cb  = (lane >> 4) * 2;

    v8f acc[2][2] = {};

    for (int k = 0; k < KT; ++k) {
        __syncthreads();                     // WAR on wLds
        // Transposing gather: w_dcn[(k,c,g),f] -> wLds[g][f][c]
#if HAS_ASYNC_LDS
        for (int i = tid; i < G_N * 32 * C_N; i += 256) {
            const int c = i & 31, f = (i >> 5) & 31, g = i >> 10;
            __builtin_amdgcn_global_load_async_to_lds_b32(
                (GLOBAL_AS int*)(w_dcn + ((k * C_N + c) * G_N + g) * (F_N / G_N) + f),
                (LDS_AS int*)&wLds[(g * 32 + f) * WSTR + c], 0, 0);
        }
        asm volatile("s_wait_asynccnt 0x0" ::: "memory");
#else
        for (int i = tid; i < G_N * 32 * C_N; i += 256) {
            const int c = i & 31, f = (i >> 5) & 31, g = i >> 10;
            wLds[(g * 32 + f) * WSTR + c] =
                w_dcn[((k * C_N + c) * G_N + g) * (F_N / G_N) + f];
        }
#endif
        __syncthreads();

        const int ky = k / 5, kx = k - 5 * ky;
        const float* ob = offs + ((b * OH_N + y) * OW_N + xp) * OFFC + k * 4;

        for (int g = 0; g < G_N; ++g) {
            // Learned offsets for (pixel, tap, group); layout (K,2,G) => k*4 + d*2 + g
            const float oy = ob[g];
            const float ox = ob[2 + g];
            float py = (float)(y + 2 * ky) + oy;   // base_y = y+4, kdy = -4+2*ky
            float px = (float)(xp + 2 * kx) + ox;
            py = fminf(fmaxf(py, 0.0f), (float)(H_N - 1));
            px = fminf(fmaxf(px, 0.0f), (float)(W_N - 1));
            const float y0f = fminf(floorf(py), (float)(H_N - 2));
            const float x0f = fminf(floorf(px), (float)(W_N - 2));
            const float wy = py - y0f, wx = px - x0f;
            const int y0i = (int)y0f, x0i = (int)x0f;
            const float w00 = (1.0f - wy) * (1.0f - wx);
            const float w01 = (1.0f - wy) * wx;
            const float w10 = wy * (1.0f - wx);
            const float w11 = wy * wx;
            const float* s00 = vol + (((b * H_N + y0i) * W_N + x0i) * C_N + h * 16);
            const float* s01 = s00 + C_N;
            const float* s10 = s00 + W_N * C_N;
            const float* s11 = s10 + C_N;

            // WAR fence: prior frag reads of aLds finish before overwrite
            asm volatile("s_wait_dscnt 0" ::: "memory");
            v4f* ap = (v4f*)&aLds[wave][p * ASTR + h * 16];
            #pragma unroll
            for (int j = 0; j < 4; ++j)
                ap[j] = ((const v4f*)s00)[j] * w00 + ((const v4f*)s01)[j] * w01 +
                        ((const v4f*)s10)[j] * w10 + ((const v4f*)s11)[j] * w11;
            // RAW fence: cross-lane panel visible before fragment loads
            asm volatile("s_wait_dscnt 0" ::: "memory");

            const float* aRow = &aLds[wave][col * ASTR];
            #pragma unroll
            for (int c0 = 0; c0 < C_N; c0 += 4) {
                v2f a = *(const v2f*)(aRow + c0 + cb);
                #pragma unroll
                for (int t = 0; t < 2; ++t) {
                    v2f bb = *(const v2f*)&wLds[(g * 32 + t * 16 + col) * WSTR + c0 + cb];
                    acc[g][t] = __builtin_amdgcn_wmma_f32_16x16x4_f32(
                        false, a, false, bb, (short)0, acc[g][t], false, false);
                }
            }
        }
    }

    const int mh = (lane >> 4) * 8;
    for (int g = 0; g < G_N; ++g)
        for (int t = 0; t < 2; ++t) {
            const int ch = g * 32 + t * 16 + col;   // (g,f) flattened to F
            const float bias = b_dcn[ch];
            #pragma unroll
            for (int i = 0; i < 8; ++i) {
                const int xm = x0 + mh + i;
                if (xm < OW_N)
                    out[((b * OH_N + y) * OW_N + xm) * F_N + ch] = acc[g][t][i] + bias;
            }
        }
}

// ---------------------------------------------------------------------------
extern "C" void kernel_launch(void* const* d_in, const int* in_sizes, int n_in,
                              void* d_out, int out_size, void* d_ws, size_t ws_size,
                              hipStream_t stream) {
    (void)in_sizes; (void)n_in; (void)out_size; (void)ws_size;
    const float* vol   = (const float*)d_in[0];
    const float* w_off = (const float*)d_in[1];
    const float* b_off = (const float*)d_in[2];
    const float* w_dcn = (const float*)d_in[3];
    const float* b_dcn = (const float*)d_in[4];
    float* out  = (float*)d_out;
    float* offs = (float*)d_ws;      // B*OH*OW*OFFC floats = 23.04 MB scratch

    dim3 grid(B_N * OH_N);           // one output row per block, 8 waves x 16 px
    offset_conv_wmma<<<grid, 256, 0, stream>>>(vol, w_off, b_off, offs);
    deform_conv_wmma<<<grid, 256, 0, stream>>>(vol, offs, w_dcn, b_dcn, out);
}